// PointNetFeaturePropagation_16758962389305
// MI455X (gfx1250) — compile-verified
//
#include <hip/hip_runtime.h>

// Problem constants (fixed by setup_inputs)
#define B_    8
#define N1_   8192
#define N2_   2048
#define C1_   128
#define C2_   256
#define CIN_  384     // C1 + C2
#define CMID_ 256
#define COUT_ 128
#define BN_EPS 1e-5f
#define MIN_DIST 1e-10f

typedef __attribute__((ext_vector_type(2))) float v2f;
typedef __attribute__((ext_vector_type(8))) float v8f;
typedef int v4i_ __attribute__((vector_size(16)));

// ---------------------------------------------------------------------------
// gfx1250 async global->LDS staging (ASYNCcnt engine), guarded for toolchains
// without the builtins.
// ---------------------------------------------------------------------------
#if defined(__HIP_DEVICE_COMPILE__)
#if __has_builtin(__builtin_amdgcn_global_load_async_to_lds_b32) && \
    __has_builtin(__builtin_amdgcn_global_load_async_to_lds_b128) && \
    __has_builtin(__builtin_amdgcn_s_wait_asynccnt)
#define FP_HAVE_ASYNC 1
#endif
#endif
#ifndef FP_HAVE_ASYNC
#define FP_HAVE_ASYNC 0
#endif

#if FP_HAVE_ASYNC
__device__ __forceinline__ void fp_async_b32(const float* g, float* l) {
  // generic->AS1 / AS3 conversions via integer casts: global generic pointers
  // are bit-identical to AS1; generic LDS pointers carry the LDS offset in the
  // low 32 bits (flat aperture rule), which is exactly the AS3 pointer value.
  __builtin_amdgcn_global_load_async_to_lds_b32(
      (__attribute__((address_space(1))) int*)(unsigned long long)g,
      (__attribute__((address_space(3))) int*)(unsigned)(unsigned long long)l,
      0, 0);
}
__device__ __forceinline__ void fp_async_b128(const float* g, float* l) {
  __builtin_amdgcn_global_load_async_to_lds_b128(
      (__attribute__((address_space(1))) v4i_*)(unsigned long long)g,
      (__attribute__((address_space(3))) v4i_*)(unsigned)(unsigned long long)l,
      0, 0);
}
#endif

// ---------------------------------------------------------------------------
// Kernel 1: 3-NN search. One thread per query point; all of xyz2[b] staged in
// LDS (2048 pts * 12B = 24KB). Writes normalized inverse-distance weights and
// neighbor indices.
// ---------------------------------------------------------------------------
__global__ void fp_knn_kernel(const float* __restrict__ xyz1,
                              const float* __restrict__ xyz2,
                              float* __restrict__ knn_w,
                              int* __restrict__ knn_idx) {
  __shared__ float sx[N2_], sy[N2_], sz[N2_];
  const int b = blockIdx.y;
  const int i = blockIdx.x * blockDim.x + threadIdx.x;

  const float* p2 = xyz2 + (size_t)b * N2_ * 3;
  for (int j = threadIdx.x; j < N2_; j += blockDim.x) {
    sx[j] = p2[j * 3 + 0];
    sy[j] = p2[j * 3 + 1];
    sz[j] = p2[j * 3 + 2];
  }
  __syncthreads();

  const size_t qoff = ((size_t)b * N1_ + i) * 3;
  const float px = xyz1[qoff + 0];
  const float py = xyz1[qoff + 1];
  const float pz = xyz1[qoff + 2];

  float d0 = 3.4e38f, d1 = 3.4e38f, d2 = 3.4e38f;
  int   i0 = 0, i1 = 0, i2 = 0;
  for (int j = 0; j < N2_; ++j) {
    const float dx = px - sx[j];
    const float dy = py - sy[j];
    const float dz = pz - sz[j];
    const float d = dx * dx + dy * dy + dz * dz;
    if (d < d0) {
      d2 = d1; i2 = i1;
      d1 = d0; i1 = i0;
      d0 = d;  i0 = j;
    } else if (d < d1) {
      d2 = d1; i2 = i1;
      d1 = d;  i1 = j;
    } else if (d < d2) {
      d2 = d;  i2 = j;
    }
  }
  float w0 = 1.0f / fmaxf(sqrtf(d0), MIN_DIST);
  float w1 = 1.0f / fmaxf(sqrtf(d1), MIN_DIST);
  float w2 = 1.0f / fmaxf(sqrtf(d2), MIN_DIST);
  const float inv = 1.0f / (w0 + w1 + w2);
  const size_t o = ((size_t)b * N1_ + i) * 3;
  knn_w[o + 0] = w0 * inv;  knn_idx[o + 0] = i0;
  knn_w[o + 1] = w1 * inv;  knn_idx[o + 1] = i1;
  knn_w[o + 2] = w2 * inv;  knn_idx[o + 2] = i2;
}

// ---------------------------------------------------------------------------
// Kernel 2: weighted feature gather. Block covers 32 points (tx) x 8 channel
// groups (ty); scattered 4B reads hit L2 (points2 is 16.8MB < 192MB L2),
// writes are coalesced along N.
// ---------------------------------------------------------------------------
__global__ void fp_gather_kernel(const float* __restrict__ points2,
                                 const float* __restrict__ knn_w,
                                 const int* __restrict__ knn_idx,
                                 float* __restrict__ interp) {
  const int b  = blockIdx.y;
  const int tx = threadIdx.x & 31;
  const int ty = threadIdx.x >> 5;   // 0..7
  const int n  = blockIdx.x * 32 + tx;

  const size_t base = ((size_t)b * N1_ + n) * 3;
  const float w0 = knn_w[base + 0];
  const float w1 = knn_w[base + 1];
  const float w2 = knn_w[base + 2];
  const int   j0 = knn_idx[base + 0];
  const int   j1 = knn_idx[base + 1];
  const int   j2 = knn_idx[base + 2];

  const float* p2 = points2 + (size_t)b * C2_ * N2_;
  for (int cc = 0; cc < C2_ / 8; ++cc) {
    const int c = ty * (C2_ / 8) + cc;
    const float* row = p2 + (size_t)c * N2_;
    interp[((size_t)b * C2_ + c) * N1_ + n] =
        w0 * row[j0] + w1 * row[j1] + w2 * row[j2];
  }
}

// ---------------------------------------------------------------------------
// Kernel 3: fp32 WMMA GEMM  Y[b] = W (MxK) * X[b] (KxN), N = 8192.
//  - X rows k <  ksplit come from Xa (channel count = ksplit)
//  - X rows k >= ksplit come from Xb (channel count = K - ksplit)
//  - APPLY_BN: x := relu(x*scale[k] + bias[k]) while staging (fuses BN+ReLU
//    of the previous layer into this GEMM's loads). BN path stages B manually;
//    otherwise B (and always A) is staged with async global->LDS when present.
// Block: 256 threads (8 waves), tile 64(M) x 128(N), BK = 32.
// Wave w owns N sub-tile w (16 cols) and all 4 M sub-tiles.
// Fragment layouts per CDNA5 ISA 7.12.2 (32-bit 16x4 A / 4x16 B / 16x16 C).
// Fragment loads are explicitly double-buffered across ks stages so the DS
// loads of stage ks+1 are in flight during the 4 WMMAs of stage ks.
// ---------------------------------------------------------------------------
template <bool APPLY_BN>
__global__ void fp_wmma_gemm_kernel(const float* __restrict__ Wmat, int M, int K,
                                    const float* __restrict__ Xa, int ksplit,
                                    const float* __restrict__ Xb,
                                    const float* __restrict__ scale,
                                    const float* __restrict__ bias,
                                    float* __restrict__ Y) {
  constexpr int BM = 64, BN = 128, BK = 32;
  __shared__ float lds_a[BM][BK + 1];   // +1 pad: stride 33 -> conflict-free M reads
  __shared__ float lds_b[BK][BN];

  const int b    = blockIdx.z;
  const int m0   = blockIdx.y * BM;
  const int n0   = blockIdx.x * BN;
  const int tid  = threadIdx.x;
  const int lane = tid & 31;
  const int wave = tid >> 5;            // 0..7 -> N sub-tile
  const int nb   = wave * 16;
  const int lrow = lane & 15;
  const int khalf = (lane < 16) ? 0 : 2;  // K split across lane halves

  v8f acc[4] = {};

  for (int kc = 0; kc < K; kc += BK) {
#if FP_HAVE_ASYNC
    // ---- A panel via async b32 lane-transfers (padded rows => 4B granule)
#pragma unroll
    for (int it = 0; it < (BM * BK) / 256; ++it) {
      const int e = it * 256 + tid;
      const int m = e >> 5;        // /32
      const int k = e & 31;
      fp_async_b32(&Wmat[(size_t)(m0 + m) * K + kc + k], &lds_a[m][k]);
    }
    if (!APPLY_BN) {
      // ---- B panel via async b128 lane-transfers (16B aligned both sides)
#pragma unroll
      for (int it = 0; it < (BK * BN / 4) / 256; ++it) {
        const int e  = it * 256 + tid;
        const int k  = e >> 5;     // BN/4 = 32 transfers per row
        const int n4 = (e & 31) * 4;
        const int gk = kc + k;     // uniform per wave (32 consecutive e per row)
        const float* src =
            (gk < ksplit)
                ? &Xa[((size_t)b * ksplit + gk) * N1_ + n0 + n4]
                : &Xb[((size_t)b * (K - ksplit) + (gk - ksplit)) * N1_ + n0 + n4];
        fp_async_b128(src, &lds_b[k][n4]);
      }
    } else {
      for (int e = tid; e < BK * BN; e += 256) {
        const int k  = e >> 7;     // /128
        const int n  = e & 127;
        const int gk = kc + k;
        float v = Xa[((size_t)b * ksplit + gk) * N1_ + n0 + n];
        v = fmaxf(v * scale[gk] + bias[gk], 0.0f);
        lds_b[k][n] = v;
      }
    }
    __builtin_amdgcn_s_wait_asynccnt(0);
#else
    // ---- Manual staging fallback
    for (int e = tid; e < BM * BK; e += 256) {
      const int m = e >> 5;
      const int k = e & 31;
      lds_a[m][k] = Wmat[(size_t)(m0 + m) * K + kc + k];
    }
    for (int e = tid; e < BK * BN; e += 256) {
      const int k  = e >> 7;
      const int n  = e & 127;
      const int gk = kc + k;
      float v;
      if (gk < ksplit)
        v = Xa[((size_t)b * ksplit + gk) * N1_ + n0 + n];
      else
        v = Xb[((size_t)b * (K - ksplit) + (gk - ksplit)) * N1_ + n0 + n];
      if (APPLY_BN)
        v = fmaxf(v * scale[gk] + bias[gk], 0.0f);
      lds_b[k][n] = v;
    }
#endif
    __syncthreads();

    // ---- Double-buffered fragment pipeline over the 8 k-steps of this panel
    v2f bf[2];
    v2f af[2][4];
    {
      const int kk = khalf;
      bf[0].x = lds_b[kk][nb + lrow];
      bf[0].y = lds_b[kk + 1][nb + lrow];
#pragma unroll
      for (int mt = 0; mt < 4; ++mt) {
        af[0][mt].x = lds_a[mt * 16 + lrow][kk];
        af[0][mt].y = lds_a[mt * 16 + lrow][kk + 1];
      }
    }
#pragma unroll
    for (int ks = 0; ks < BK / 4; ++ks) {
      const int cur = ks & 1;
      const int nxt = cur ^ 1;
      if (ks + 1 < BK / 4) {
        const int kk = (ks + 1) * 4 + khalf;
        bf[nxt].x = lds_b[kk][nb + lrow];
        bf[nxt].y = lds_b[kk + 1][nb + lrow];
#pragma unroll
        for (int mt = 0; mt < 4; ++mt) {
          af[nxt][mt].x = lds_a[mt * 16 + lrow][kk];
          af[nxt][mt].y = lds_a[mt * 16 + lrow][kk + 1];
        }
      }
#pragma unroll
      for (int mt = 0; mt < 4; ++mt) {
        acc[mt] = __builtin_amdgcn_wmma_f32_16x16x4_f32(
            false, af[cur][mt], false, bf[cur], (short)0, acc[mt], false, false);
      }
    }
    __syncthreads();
  }

  // C/D layout: VGPR j -> M = j (lanes 0-15) / j+8 (lanes 16-31), N = lane%16
  const int n = n0 + nb + lrow;
  const int mhi = (lane < 16) ? 0 : 8;
#pragma unroll
  for (int mt = 0; mt < 4; ++mt) {
#pragma unroll
    for (int j = 0; j < 8; ++j) {
      const int m = m0 + mt * 16 + j + mhi;
      Y[((size_t)b * M + m) * N1_ + n] = acc[mt][j];
    }
  }
}

// ---------------------------------------------------------------------------
// Kernel 4: deterministic per-channel BN stats (training-mode, biased var over
// B*N). One block per channel, fixed-order tree reduction -> scale/bias such
// that yn = y*scale + bias. (Conv bias cancels in BN, so it is never added.)
// ---------------------------------------------------------------------------
__global__ void fp_bn_stats_kernel(const float* __restrict__ Y, int C,
                                   const float* __restrict__ gamma,
                                   const float* __restrict__ beta,
                                   float* __restrict__ scale,
                                   float* __restrict__ bias) {
  __shared__ float ssum[256], ssq[256];
  const int c   = blockIdx.x;
  const int tid = threadIdx.x;
  float s = 0.0f, q = 0.0f;
  for (int b = 0; b < B_; ++b) {
    const float* row = Y + ((size_t)b * C + c) * N1_;
    for (int n = tid; n < N1_; n += 256) {
      const float v = row[n];
      s += v;
      q += v * v;
    }
  }
  ssum[tid] = s;
  ssq[tid]  = q;
  __syncthreads();
  for (int off = 128; off > 0; off >>= 1) {
    if (tid < off) {
      ssum[tid] += ssum[tid + off];
      ssq[tid]  += ssq[tid + off];
    }
    __syncthreads();
  }
  if (tid == 0) {
    const float inv = 1.0f / (float)(B_ * N1_);
    const float mu  = ssum[0] * inv;
    const float var = ssq[0] * inv - mu * mu;
    const float sc  = gamma[c] * rsqrtf(var + BN_EPS);
    scale[c] = sc;
    bias[c]  = beta[c] - mu * sc;
  }
}

// ---------------------------------------------------------------------------
// Kernel 5: final in-place BN + ReLU on d_out.
// ---------------------------------------------------------------------------
__global__ void fp_bn_apply_kernel(float* __restrict__ Y,
                                   const float* __restrict__ scale,
                                   const float* __restrict__ bias, int C) {
  const int idx = blockIdx.x * blockDim.x + threadIdx.x;
  const int c = (idx / N1_) % C;
  Y[idx] = fmaxf(Y[idx] * scale[c] + bias[c], 0.0f);
}

// ---------------------------------------------------------------------------
extern "C" void kernel_launch(void* const* d_in, const int* in_sizes, int n_in,
                              void* d_out, int out_size, void* d_ws, size_t ws_size,
                              hipStream_t stream) {
  (void)in_sizes; (void)n_in; (void)out_size; (void)ws_size;

  const float* xyz1    = (const float*)d_in[0];
  const float* xyz2    = (const float*)d_in[1];
  const float* points1 = (const float*)d_in[2];
  const float* points2 = (const float*)d_in[3];
  const float* w0      = (const float*)d_in[4];
  // d_in[5] = b0  (unused: conv bias cancels in training-mode BN)
  const float* gamma0  = (const float*)d_in[6];
  const float* beta0   = (const float*)d_in[7];
  const float* w1      = (const float*)d_in[8];
  // d_in[9] = b1  (unused)
  const float* gamma1  = (const float*)d_in[10];
  const float* beta1   = (const float*)d_in[11];
  float* out = (float*)d_out;

  // Workspace carve-up
  char* w = (char*)d_ws;
  int*   knn_idx = (int*)w;    w += (size_t)B_ * N1_ * 3 * sizeof(int);
  float* knn_w   = (float*)w;  w += (size_t)B_ * N1_ * 3 * sizeof(float);
  float* interp  = (float*)w;  w += (size_t)B_ * C2_ * N1_ * sizeof(float);
  float* y0      = (float*)w;  w += (size_t)B_ * CMID_ * N1_ * sizeof(float);
  float* scale0  = (float*)w;  w += CMID_ * sizeof(float);
  float* bias0   = (float*)w;  w += CMID_ * sizeof(float);
  float* scale1  = (float*)w;  w += COUT_ * sizeof(float);
  float* bias1   = (float*)w;  w += COUT_ * sizeof(float);

  // 1) 3-NN + inverse-distance weights
  fp_knn_kernel<<<dim3(N1_ / 256, B_), 256, 0, stream>>>(xyz1, xyz2, knn_w, knn_idx);

  // 2) Weighted gather of points2 -> interp [B, C2, N1]
  fp_gather_kernel<<<dim3(N1_ / 32, B_), 256, 0, stream>>>(points2, knn_w, knn_idx, interp);

  // 3) Layer 0 GEMM: y0 = W0 @ concat(points1, interp)   (256x384 * 384x8192)
  fp_wmma_gemm_kernel<false><<<dim3(N1_ / 128, CMID_ / 64, B_), 256, 0, stream>>>(
      w0, CMID_, CIN_, points1, C1_, interp, nullptr, nullptr, y0);

  // 4) BN0 stats -> per-channel scale/bias
  fp_bn_stats_kernel<<<dim3(CMID_), 256, 0, stream>>>(y0, CMID_, gamma0, beta0, scale0, bias0);

  // 5) Layer 1 GEMM with fused BN0+ReLU on loads: out = W1 @ relu(bn0(y0))
  fp_wmma_gemm_kernel<true><<<dim3(N1_ / 128, COUT_ / 64, B_), 256, 0, stream>>>(
      w1, COUT_, CMID_, y0, CMID_, nullptr, scale0, bias0, out);

  // 6) BN1 stats
  fp_bn_stats_kernel<<<dim3(COUT_), 256, 0, stream>>>(out, COUT_, gamma1, beta1, scale1, bias1);

  // 7) Final in-place BN1 + ReLU
  fp_bn_apply_kernel<<<dim3((B_ * COUT_ * N1_) / 256), 256, 0, stream>>>(out, scale1, bias1, COUT_);
}